// RGCN_8701603741709
// MI455X (gfx1250) — compile-verified
//
#include <hip/hip_runtime.h>
#include <math.h>

typedef __attribute__((ext_vector_type(2))) float v2f;
typedef __attribute__((ext_vector_type(8))) float v8f;

#define N_NODES 100000
#define N_EDGES 500000
#define N_REL   3

// ---------------------------------------------------------------------------
// Utility kernels
// ---------------------------------------------------------------------------
__global__ __launch_bounds__(256)
void zero_kernel(float* __restrict__ p, int n) {
    int i = blockIdx.x * 256 + threadIdx.x;
    if (i < n) p[i] = 0.0f;
}

__global__ __launch_bounds__(1)
void set_u32(unsigned* __restrict__ p, unsigned v) { p[0] = v; }

// Float histogram of src -> out-degree per relation (runs once, tiny).
__global__ __launch_bounds__(256)
void deg_out_count(const int* __restrict__ src, float* __restrict__ degO, int total) {
    int i = blockIdx.x * 256 + threadIdx.x;
    if (i >= total) return;
    int r = i / N_EDGES;
    __hip_atomic_fetch_add(&degO[(size_t)r * N_NODES + src[i]], 1.0f,
                           __ATOMIC_RELAXED, __HIP_MEMORY_SCOPE_AGENT);
}

// In-place deg -> rsqrt(max(deg,1))
__global__ __launch_bounds__(256)
void rsqrt_clamp(float* __restrict__ p, int n) {
    int i = blockIdx.x * 256 + threadIdx.x;
    if (i < n) p[i] = rsqrtf(fmaxf(p[i], 1.0f));
}

// Row-wise L2 normalize x [N,128] -> h. One block (128 threads) per row.
__global__ __launch_bounds__(128)
void l2norm(const float* __restrict__ x, float* __restrict__ h) {
    __shared__ float red[4];
    int row = blockIdx.x;
    int t = threadIdx.x;
    float v = x[(size_t)row * 128 + t];
    float ss = v * v;
    #pragma unroll
    for (int off = 16; off > 0; off >>= 1)
        ss += __shfl_xor(ss, off, 32);
    if ((t & 31) == 0) red[t >> 5] = ss;
    __syncthreads();
    float tot = red[0] + red[1] + red[2] + red[3];
    float scale = 1.0f / fmaxf(sqrtf(tot), 1e-12f);
    h[(size_t)row * 128 + t] = v * scale;
}

// ---------------------------------------------------------------------------
// CSC construction (per relation): histogram -> exclusive scan -> fill
// ---------------------------------------------------------------------------
__global__ __launch_bounds__(256)
void hist_u32(const int* __restrict__ idx, unsigned* __restrict__ cnt, int n) {
    int i = blockIdx.x * 256 + threadIdx.x;
    if (i < n) atomicAdd(&cnt[idx[i]], 1u);
}

// Block-wise exclusive scan (Hillis-Steele in LDS) + per-block totals.
__global__ __launch_bounds__(256)
void scan_block(const unsigned* __restrict__ cnt, unsigned* __restrict__ excl,
                unsigned* __restrict__ partials, int n) {
    __shared__ unsigned sh[256];
    int tid = threadIdx.x;
    int i = blockIdx.x * 256 + tid;
    unsigned v = (i < n) ? cnt[i] : 0u;
    sh[tid] = v;
    __syncthreads();
    #pragma unroll
    for (int off = 1; off < 256; off <<= 1) {
        unsigned t = (tid >= off) ? sh[tid - off] : 0u;
        __syncthreads();
        sh[tid] += t;
        __syncthreads();
    }
    unsigned incl = sh[tid];
    if (i < n) excl[i] = incl - v;
    if (tid == 255) partials[blockIdx.x] = incl;
}

// Serial exclusive scan of block partials (nb ~ 391, trivial).
__global__ __launch_bounds__(1)
void scan_partials(unsigned* __restrict__ partials, int nb) {
    unsigned run = 0;
    for (int b = 0; b < nb; ++b) {
        unsigned t = partials[b];
        partials[b] = run;
        run += t;
    }
}

__global__ __launch_bounds__(256)
void scan_add(unsigned* __restrict__ excl, const unsigned* __restrict__ partials, int n) {
    int i = blockIdx.x * 256 + threadIdx.x;
    if (i < n) excl[i] += partials[blockIdx.x];
}

// rsI[d] = rsqrt(max(in_degree, 1)) from row_ptr diffs.
__global__ __launch_bounds__(256)
void rsqrt_from_rowptr(const unsigned* __restrict__ rp, float* __restrict__ rsI, int n) {
    int i = blockIdx.x * 256 + threadIdx.x;
    if (i < n) {
        float d = (float)(rp[i + 1] - rp[i]);
        rsI[i] = rsqrtf(fmaxf(d, 1.0f));
    }
}

// Scatter edges into CSC order: col[row_ptr[dst]+k] = src (atomic cursor).
__global__ __launch_bounds__(256)
void fill_csr(const int* __restrict__ dst, const int* __restrict__ src,
              const unsigned* __restrict__ row_ptr, unsigned* __restrict__ cursor,
              int* __restrict__ col, int n) {
    int i = blockIdx.x * 256 + threadIdx.x;
    if (i >= n) return;
    int d = dst[i];
    unsigned pos = row_ptr[d] + atomicAdd(&cursor[d], 1u);
    col[pos] = src[i];
}

// ---------------------------------------------------------------------------
// Dense part: m = (h * rsqrt(deg_out)) @ W   via V_WMMA_F32_16X16X4_F32
// Block = 128 threads = 4 waves; block tile 16 rows x 64 cols;
// each wave one 16x16 tile; grid = (N/16, Dout/64). Din fixed at 128.
// ---------------------------------------------------------------------------
__global__ __launch_bounds__(128)
void gemm_wmma(const float* __restrict__ h, const float* __restrict__ rsO,
               const float* __restrict__ W, float* __restrict__ m, int Dout) {
    const int DIN = 128;
    const int lane    = threadIdx.x & 31;
    const int wave    = threadIdx.x >> 5;
    const int rowBase = blockIdx.x * 16;
    const int colBase = blockIdx.y * 64 + wave * 16;
    const int halfSel = lane >> 4;
    const int mIdx    = lane & 15;
    const int kb      = halfSel * 2;

    const int aRow = rowBase + mIdx;
    const float s  = rsO[aRow];
    const float* aPtr = h + (size_t)aRow * DIN + kb;
    const float* bPtr = W + (size_t)kb * Dout + colBase + mIdx;

    v8f acc = {};
    #pragma unroll 8
    for (int k = 0; k < DIN; k += 4) {
        v2f a;
        a.x = aPtr[k]     * s;
        a.y = aPtr[k + 1] * s;
        v2f b;
        b.x = bPtr[(size_t)k * Dout];
        b.y = bPtr[(size_t)(k + 1) * Dout];
        acc = __builtin_amdgcn_wmma_f32_16x16x4_f32(
            false, a, false, b, (short)0, acc, false, false);
    }

    #pragma unroll
    for (int v = 0; v < 8; ++v) {
        int row = rowBase + v + halfSel * 8;
        m[(size_t)row * Dout + colBase + mIdx] = acc[v];
    }
}

// ---------------------------------------------------------------------------
// Sparse part (pull model): one wave per dst node, register accumulation,
// no atomics. Each lane owns F/32 consecutive floats (b128/b64 loads).
// ---------------------------------------------------------------------------
template <int F>
__global__ __launch_bounds__(256)
void spmm_gather(const float* __restrict__ m, const unsigned* __restrict__ row_ptr,
                 const int* __restrict__ col, const float* __restrict__ rsI,
                 float* __restrict__ hout) {
    constexpr int V = F / 32;
    const int wave = threadIdx.x >> 5;
    const int lane = threadIdx.x & 31;
    const int d = blockIdx.x * 8 + wave;
    if (d >= N_NODES) return;

    const unsigned beg = row_ptr[d];
    const unsigned end = row_ptr[d + 1];
    const int fb = lane * V;

    float acc[V];
    #pragma unroll
    for (int j = 0; j < V; ++j) acc[j] = 0.0f;

    for (unsigned t = beg; t < end; ++t) {
        int s = col[t];
        if (t + 1 < end)  // hide gather latency: prefetch next source row
            __builtin_prefetch(m + (size_t)col[t + 1] * F + fb, 0, 0);
        const float* mr = m + (size_t)s * F + fb;
        #pragma unroll
        for (int j = 0; j < V; ++j) acc[j] += mr[j];
    }

    const float w = rsI[d];
    float* o = hout + (size_t)d * F + fb;
    #pragma unroll
    for (int j = 0; j < V; ++j) o[j] += acc[j] * w;
}

// out = hin + sum_r b[r]  (+ relu)
__global__ __launch_bounds__(256)
void finalize(const float* __restrict__ hin, const float* __restrict__ b,
              float* __restrict__ out, int F, int do_relu, int total) {
    int i = blockIdx.x * 256 + threadIdx.x;
    if (i >= total) return;
    int f = i % F;
    float v = hin[i] + b[f] + b[F + f] + b[2 * F + f];
    out[i] = do_relu ? fmaxf(v, 0.0f) : v;
}

// ---------------------------------------------------------------------------
// Host-side orchestration
// ---------------------------------------------------------------------------
extern "C" void kernel_launch(void* const* d_in, const int* in_sizes, int n_in,
                              void* d_out, int out_size, void* d_ws, size_t ws_size,
                              hipStream_t stream) {
    (void)in_sizes; (void)n_in; (void)out_size; (void)ws_size;

    const float* x   = (const float*)d_in[0];
    const int*   src = (const int*)d_in[1];
    const int*   dst = (const int*)d_in[2];
    const float* Wl[4] = {(const float*)d_in[3], (const float*)d_in[5],
                          (const float*)d_in[7], (const float*)d_in[9]};
    const float* Bl[4] = {(const float*)d_in[4], (const float*)d_in[6],
                          (const float*)d_in[8], (const float*)d_in[10]};
    float* out = (float*)d_out;

    const int N = N_NODES, E = N_EDGES, R = N_REL;

    // Scratch layout:
    // floats: [degO R*N][rsI R*N][h0 N*128][h1 N*128][m N*128]
    // u32/int: [row_ptr R*(N+1)][col R*E][counts N][partials 512]
    float* ws   = (float*)d_ws;
    float* degO = ws;
    float* rsI  = degO + (size_t)R * N;
    float* h0   = rsI + (size_t)R * N;
    float* h1   = h0 + (size_t)N * 128;
    float* mbuf = h1 + (size_t)N * 128;
    unsigned* row_ptr = (unsigned*)(mbuf + (size_t)N * 128);
    int*      col     = (int*)(row_ptr + (size_t)R * (N + 1));
    unsigned* counts  = (unsigned*)(col + (size_t)R * E);
    unsigned* partials = counts + N;

    const int nbN = (N + 255) / 256;   // 391 blocks over N

    // Out-degree scaling factors (shared across all layers)
    zero_kernel<<<(R * N + 255) / 256, 256, 0, stream>>>(degO, R * N);
    deg_out_count<<<(R * E + 255) / 256, 256, 0, stream>>>(src, degO, R * E);
    rsqrt_clamp<<<(R * N + 255) / 256, 256, 0, stream>>>(degO, R * N);

    // h0 = L2-normalized x
    l2norm<<<N, 128, 0, stream>>>(x, h0);

    // CSC per relation: row_ptr_r, col_r, rsI_r
    for (int r = 0; r < R; ++r) {
        unsigned* rp = row_ptr + (size_t)r * (N + 1);
        int* cl = col + (size_t)r * E;
        const int* dr = dst + (size_t)r * E;
        const int* sr = src + (size_t)r * E;

        zero_kernel<<<nbN, 256, 0, stream>>>((float*)counts, N);
        hist_u32<<<(E + 255) / 256, 256, 0, stream>>>(dr, counts, E);
        scan_block<<<nbN, 256, 0, stream>>>(counts, rp, partials, N);
        scan_partials<<<1, 1, 0, stream>>>(partials, nbN);
        scan_add<<<nbN, 256, 0, stream>>>(rp, partials, N);
        set_u32<<<1, 1, 0, stream>>>(rp + N, (unsigned)E);
        rsqrt_from_rowptr<<<nbN, 256, 0, stream>>>(rp, rsI + (size_t)r * N, N);
        zero_kernel<<<nbN, 256, 0, stream>>>((float*)counts, N);  // cursor
        fill_csr<<<(E + 255) / 256, 256, 0, stream>>>(dr, sr, rp, counts, cl, E);
    }

    // Layers
    float* hin   = h0;
    float* hnext = h1;
    for (int l = 0; l < 4; ++l) {
        const int Fout = (l == 3) ? 64 : 128;
        const int tot  = N * Fout;

        zero_kernel<<<(tot + 255) / 256, 256, 0, stream>>>(hnext, tot);

        for (int r = 0; r < R; ++r) {
            dim3 grid(N / 16, Fout / 64);
            gemm_wmma<<<grid, 128, 0, stream>>>(
                hin, degO + (size_t)r * N,
                Wl[l] + (size_t)r * 128 * Fout, mbuf, Fout);

            unsigned* rp = row_ptr + (size_t)r * (N + 1);
            int* cl = col + (size_t)r * E;
            if (Fout == 128)
                spmm_gather<128><<<(N + 7) / 8, 256, 0, stream>>>(
                    mbuf, rp, cl, rsI + (size_t)r * N, hnext);
            else
                spmm_gather<64><<<(N + 7) / 8, 256, 0, stream>>>(
                    mbuf, rp, cl, rsI + (size_t)r * N, hnext);
        }

        float* dest = (l == 3) ? out : hnext;
        finalize<<<(tot + 255) / 256, 256, 0, stream>>>(
            hnext, Bl[l], dest, Fout, (l != 3) ? 1 : 0, tot);

        float* tmp = hin; hin = hnext; hnext = tmp;
    }
}